// NeRFRenderer_2937757630585
// MI455X (gfx1250) — compile-verified
//
#include <hip/hip_runtime.h>

#define GRID    128
#define NCELLS  (GRID * GRID * GRID)   // 2,097,152
#define NCASC   3

typedef __attribute__((ext_vector_type(2))) float v2f;
typedef __attribute__((ext_vector_type(8))) float v8f;

// ---------------------------------------------------------------------------
// Morton decode (inverse of _part1by2 in the reference)
// ---------------------------------------------------------------------------
__device__ __forceinline__ unsigned compact1by2(unsigned x) {
    x &= 0x09249249u;
    x = (x ^ (x >> 2))  & 0x030c30c3u;
    x = (x ^ (x >> 4))  & 0x0300f00fu;
    x = (x ^ (x >> 8))  & 0xff0000ffu;
    x = (x ^ (x >> 16)) & 0x000003ffu;
    return x;
}

// ---------------------------------------------------------------------------
// Setup: build augmented B operands in V_WMMA_F32_16X16X4_F32 lane layout.
//   B is 4x16 (K x N), N = camera. Rows K=0..2 hold R[cam][k][comp]; the
//   K=3 row holds -(t@R)[cam][comp] so the WMMA computes the full affine
//   transform with C = inline 0 (A's K=3 element is 1.0).
//   Lane layout: lanes0-15 {V0=K0, V1=K1}, lanes16-31 {V0=K2, V1=K3}.
//   ws[((comp*2+half)*32 + lane)*2 + v] : B operand values (384 floats)
//   ws[384] = cx/fx, ws[385] = cy/fy
// ---------------------------------------------------------------------------
__global__ void nerf_setup_kernel(const float* __restrict__ poses,
                                  const float* __restrict__ intr,
                                  float* __restrict__ ws) {
    const int  lane = threadIdx.x;         // 32 threads, one wave
    const int  lo   = lane & 15;
    const bool low  = lane < 16;
    for (int comp = 0; comp < 3; ++comp) {
        for (int half = 0; half < 2; ++half) {
            const int cam = half * 16 + lo;
            float ntr = 0.0f;              // -(t @ R)[cam][comp]
            for (int j = 0; j < 3; ++j)
                ntr -= poses[cam * 16 + j * 4 + 3] * poses[cam * 16 + j * 4 + comp];
            for (int v = 0; v < 2; ++v) {
                const int k = low ? v : v + 2;          // K row of B
                const float val = (k < 3) ? poses[cam * 16 + k * 4 + comp] : ntr;
                ws[((comp * 2 + half) * 32 + lane) * 2 + v] = val;
            }
        }
    }
    if (lane == 0) {
        ws[384] = intr[2] / intr[0];       // cx/fx
        ws[385] = intr[3] / intr[1];       // cy/fy
    }
}

// ---------------------------------------------------------------------------
// Main kernel: each wave handles 32 consecutive Morton codes (2 WMMA M-tiles).
// Per cascade: 2 tiles x 2 cam-halves x 3 comps = 12 WMMAs (C = inline 0).
// Per half-row: 2 v_fmaak thresholds + 3 single-v_cmp ballots (abs via free
// source modifiers), ANDed on the scalar pipe; cam-halves merged before the
// one bit-test chain per row pair.
// ---------------------------------------------------------------------------
__global__ __launch_bounds__(256) void nerf_mark_kernel(
        const float* __restrict__ density,
        const float* __restrict__ ws,
        float* __restrict__ out) {
    const int      lane = threadIdx.x & 31;
    const unsigned wave = (blockIdx.x * blockDim.x + threadIdx.x) >> 5;
    const unsigned base = wave << 5;
    const unsigned m    = base + lane;      // this lane's Morton code / cell

    // Decode Morton -> grid coords -> world coords in [-1, 1]
    const unsigned xi = compact1by2(m);
    const unsigned yi = compact1by2(m >> 1);
    const unsigned zi = compact1by2(m >> 2);
    const float inv = 2.0f / (float)(GRID - 1);
    const float wx  = fmaf((float)xi, inv, -1.0f);
    const float wy  = fmaf((float)yi, inv, -1.0f);
    const float wz  = fmaf((float)zi, inv, -1.0f);

    const float sx = ws[384];               // cx/fx, uniform -> s_load
    const float sy = ws[385];               // cy/fy

    // Augmented B operands (one b64 load per (comp,half))
    v2f B[3][2];
#pragma unroll
    for (int comp = 0; comp < 3; ++comp)
#pragma unroll
        for (int half = 0; half < 2; ++half)
            B[comp][half] = *(const v2f*)&ws[((comp * 2 + half) * 32 + lane) * 2];

    // Unscaled augmented A operands for the two 16-cell tiles.
    // lanes0-15: V0=K0(wx), V1=K1(wy); lanes16-31: V0=K2(wz), V1=K3(1.0)
    const int   src  = lane & 15;
    const float wzT0 = __shfl(wz, src, 32);
    const float wxT1 = __shfl(wx, src + 16, 32);
    const float wyT1 = __shfl(wy, src + 16, 32);
    const bool  low  = lane < 16;
    float a0[2], a1[2];
    a0[0] = low ? wx   : wzT0;  a1[0] = low ? wy   : 1.0f;
    a0[1] = low ? wxT1 : wz;    a1[1] = low ? wyT1 : 1.0f;

    // cascade: bnd = min(2^cas,4), h = bnd/GRID, scale = bnd-h, th = 2h
    const float scl[NCASC] = {1.0f - 1.0f / GRID, 2.0f - 2.0f / GRID, 4.0f - 4.0f / GRID};
    const float th2[NCASC] = {2.0f / GRID, 4.0f / GRID, 8.0f / GRID};
    // in_aabb is identically true (|world*(bnd-h)| <= bnd-h < BOUND+h),
    // so untrained == !visible.

    const v8f zeroC = {};                   // -> inline SRC2 = 0

#pragma unroll
    for (int cas = 0; cas < NCASC; ++cas) {
        const float s   = scl[cas];
        const float th  = th2[cas];
        const float sv  = low ? s : 1.0f;   // K3 row (==1.0) must stay unscaled
        unsigned vis[2] = {0u, 0u};         // per-tile 16-bit "any camera sees row"
#pragma unroll
        for (int tile = 0; tile < 2; ++tile) {
            v2f A;
            A.x = a0[tile] * s;             // K0/K2 scaled by cascade
            A.y = a1[tile] * sv;            // K1 scaled, K3 stays 1.0

            // D[M=cell, N=cam] = world_s @ R - t@R  (affine via K=3 row)
            v8f X0 = __builtin_amdgcn_wmma_f32_16x16x4_f32(
                         false, A, false, B[0][0], (short)0, zeroC, false, false);
            v8f Y0 = __builtin_amdgcn_wmma_f32_16x16x4_f32(
                         false, A, false, B[1][0], (short)0, zeroC, false, false);
            v8f Z0 = __builtin_amdgcn_wmma_f32_16x16x4_f32(
                         false, A, false, B[2][0], (short)0, zeroC, false, false);
            v8f X1 = __builtin_amdgcn_wmma_f32_16x16x4_f32(
                         false, A, false, B[0][1], (short)0, zeroC, false, false);
            v8f Y1 = __builtin_amdgcn_wmma_f32_16x16x4_f32(
                         false, A, false, B[1][1], (short)0, zeroC, false, false);
            v8f Z1 = __builtin_amdgcn_wmma_f32_16x16x4_f32(
                         false, A, false, B[2][1], (short)0, zeroC, false, false);

#pragma unroll
            for (int r = 0; r < 8; ++r) {
                // half 0: one v_cmp per ballot (abs is a free src modifier)
                const float z0  = Z0[r];
                const float tx0 = fmaf(sx, z0, th);
                const float ty0 = fmaf(sy, z0, th);
                const unsigned c0 = (unsigned)__ballot(z0 > 0.0f)
                                  & (unsigned)__ballot(__builtin_fabsf(X0[r]) < tx0)
                                  & (unsigned)__ballot(__builtin_fabsf(Y0[r]) < ty0);
                // half 1
                const float z1  = Z1[r];
                const float tx1 = fmaf(sx, z1, th);
                const float ty1 = fmaf(sy, z1, th);
                const unsigned c1 = (unsigned)__ballot(z1 > 0.0f)
                                  & (unsigned)__ballot(__builtin_fabsf(X1[r]) < tx1)
                                  & (unsigned)__ballot(__builtin_fabsf(Y1[r]) < ty1);
                // merge cam-halves, then one bit-test chain per row pair
                const unsigned b = c0 | c1;
                vis[tile] |= ((b & 0xffffu) ? 1u : 0u) << r;        // row M=r
                vis[tile] |= ((b >> 16)     ? 1u : 0u) << (r + 8);  // row M=r+8
            }
        }
        // lane l (tile = l>>4, row = l&15) writes its own cell
        const unsigned vmask   = low ? vis[0] : vis[1];
        const bool     visible = (vmask >> (lane & 15)) & 1u;
        const unsigned idx     = (unsigned)cas * (unsigned)NCELLS + m;
        out[idx] = visible ? density[idx] : -1.0f;
    }
}

extern "C" void kernel_launch(void* const* d_in, const int* in_sizes, int n_in,
                              void* d_out, int out_size, void* d_ws, size_t ws_size,
                              hipStream_t stream) {
    const float* density = (const float*)d_in[0];   // (3, 128^3) f32
    const float* poses   = (const float*)d_in[1];   // (32, 4, 4) f32
    const float* intr    = (const float*)d_in[2];   // (4,)       f32
    float*       out     = (float*)d_out;           // (3, 128^3) f32
    float*       ws      = (float*)d_ws;            // needs 386 floats

    nerf_setup_kernel<<<1, 32, 0, stream>>>(poses, intr, ws);

    // 2^21 cells, 32 cells per wave, 8 waves per 256-thread block -> 8192 blocks
    nerf_mark_kernel<<<NCELLS / 256, 256, 0, stream>>>(density, ws, out);
}